// FeatureFusionModel_17351667876588
// MI455X (gfx1250) — compile-verified
//
#include <hip/hip_runtime.h>

typedef __attribute__((ext_vector_type(2))) float v2f;
typedef __attribute__((ext_vector_type(8))) float v8f;

#define Bn    2
#define Cn    6
#define Vn    50000
#define Pn    100000
#define Dn    384
#define Fn    64
#define K1    448
#define N1    256
#define N2    64
#define N3    16
#define Mtok  1369
#define GRIDP 37

// Padded LDS leading dims: stride % 64 banks == 4 -> 16 rows map to 16 distinct banks
#define XS_LD 452
#define H1_LD 260
#define H2_LD 68

// One wave (wave32) per block; each wave owns a 16-voxel-row tile.
__global__ __launch_bounds__(32) void fused_proj_mlp_kernel(
    const float* __restrict__ patch_tokens,     // (B,C,1369,384)
    const float* __restrict__ voxel_features,   // (B,V,64)
    const float* __restrict__ voxel_coords,     // (B,V,3)
    const float* __restrict__ image_sizes,      // (B,2)  [H,W]
    const float* __restrict__ cam_intrinsics,   // (B,C,3,3)
    const float* __restrict__ lidar2cam,        // (B,C,4,4)
    const float* __restrict__ W1, const float* __restrict__ b1,   // (448,256),(256)
    const float* __restrict__ W2, const float* __restrict__ b2,   // (256,64),(64)
    const float* __restrict__ W3, const float* __restrict__ b3,   // (64,16),(16)
    float* __restrict__ voxel_scores)           // (B,V,16) scratch
{
    __shared__ float Xs [16 * XS_LD];   // fused features tile 16x448
    __shared__ float H1s[16 * H1_LD];   // layer-1 activations 16x256
    __shared__ float H2s[16 * H2_LD];   // layer-2 activations 16x64
    __shared__ int   sflat[16 * Cn];    // per (row,cam) flat patch index
    __shared__ float sval [16 * Cn];    // per (row,cam) validity mask

    const int lane  = threadIdx.x;
    const int rbase = blockIdx.x * 16;  // global fused-row base over B*V (exact multiple)

    // ---------------- Stage 1: camera projection (16 rows x 6 cams = 96 tasks) ----
    for (int t = lane; t < 16 * Cn; t += 32) {
        const int row = t / Cn;
        const int c   = t - row * Cn;
        const int r   = rbase + row;
        const int b   = r / Vn;
        const int v   = r - b * Vn;

        const float x  = voxel_coords[(b * Vn + v) * 3 + 0];
        const float y  = voxel_coords[(b * Vn + v) * 3 + 1];
        const float zc = voxel_coords[(b * Vn + v) * 3 + 2];

        const float* E = lidar2cam + (b * Cn + c) * 16;       // 4x4 row-major
        const float cx = E[0]*x + E[1]*y + E[ 2]*zc + E[ 3];
        const float cy = E[4]*x + E[5]*y + E[ 6]*zc + E[ 7];
        const float cz = E[8]*x + E[9]*y + E[10]*zc + E[11];

        const float* Km = cam_intrinsics + (b * Cn + c) * 9;  // 3x3 row-major
        const float p0 = Km[0]*cx + Km[1]*cy + Km[2]*cz;
        const float p1 = Km[3]*cx + Km[4]*cy + Km[5]*cz;
        const float p2 = Km[6]*cx + Km[7]*cy + Km[8]*cz;

        const float denom = fmaxf(p2, 1e-12f);
        const float u  = p0 / denom;
        const float vv = p1 / denom;

        const float Hsz = image_sizes[b * 2 + 0];
        const float Wsz = image_sizes[b * 2 + 1];
        const bool valid = (cz > 0.f) && (u >= 0.f) && (u < Wsz) &&
                           (vv >= 0.f) && (vv < Hsz);

        float cu = valid ? u  : -1.f;
        float cv = valid ? vv : -1.f;
        cu *= 518.f / fmaxf(Wsz, 1e-6f);
        cv *= 518.f / fmaxf(Hsz, 1e-6f);

        int pxi = (int)(cu / 14.f);   // trunc-toward-zero, matches astype(int32)
        int pyi = (int)(cv / 14.f);
        pxi = min(max(pxi, 0), GRIDP - 1);
        pyi = min(max(pyi, 0), GRIDP - 1);
        const int flat = min(max(pyi * GRIDP + pxi, 0), Mtok - 1);

        sflat[t] = flat;
        sval [t] = valid ? 1.f : 0.f;
    }
    __syncthreads();

    // ---------------- Stage 2: build fused X tile (16 x 448) in LDS -------------
    // First 64 cols: voxel_features copy
    for (int t = lane; t < 16 * Fn; t += 32) {
        const int row = t >> 6;
        const int f   = t & 63;
        const int r   = rbase + row;
        const int b   = r / Vn;
        const int v   = r - b * Vn;
        Xs[row * XS_LD + f] = voxel_features[(b * Vn + v) * Fn + f];
    }
    // Next 384 cols: masked camera-average of gathered patch tokens (L2-resident)
    for (int t = lane; t < 16 * Dn; t += 32) {
        const int row = t / Dn;
        const int d   = t - row * Dn;
        const int r   = rbase + row;
        const int b   = r / Vn;
        float acc = 0.f, cnt = 0.f;
        #pragma unroll
        for (int c = 0; c < Cn; ++c) {
            const float m    = sval [row * Cn + c];
            const int   flat = sflat[row * Cn + c];
            const float g = patch_tokens[((size_t)(b * Cn + c) * Mtok + flat) * Dn + d];
            acc += g * m;
            cnt += m;
        }
        Xs[row * XS_LD + Fn + d] = acc / fmaxf(cnt, 1.f);
    }
    __syncthreads();

    // WMMA f32 16x16x4 operand mapping (wave32):
    //   A (16x4): row = lane&15 ; K = (lane>=16 ? 2 : 0) + vgpr
    //   B (4x16): col = lane&15 ; K = (lane>=16 ? 2 : 0) + vgpr
    //   C/D (16x16): N = lane&15 ; M = vgpr + 8*(lane>=16)
    const int lrow  = lane & 15;
    const int khalf = (lane >> 4) << 1;  // 0 or 2
    const int moff  = (lane >> 4) << 3;  // 0 or 8

    // ---------------- Layer 1: [16x448] @ W1[448x256] + b1, ReLU -> H1s ---------
    for (int nt = 0; nt < N1 / 16; ++nt) {
        const int ncol = nt * 16 + lrow;
        v8f acc = {0.f, 0.f, 0.f, 0.f, 0.f, 0.f, 0.f, 0.f};
        for (int kk = 0; kk < K1; kk += 4) {
            v2f a, bm;
            a.x  = Xs[lrow * XS_LD + kk + khalf];
            a.y  = Xs[lrow * XS_LD + kk + khalf + 1];
            bm.x = W1[(kk + khalf)     * N1 + ncol];
            bm.y = W1[(kk + khalf + 1) * N1 + ncol];
            acc = __builtin_amdgcn_wmma_f32_16x16x4_f32(
                false, a, false, bm, (short)0, acc, false, false);
        }
        const float bias = b1[ncol];
        #pragma unroll
        for (int j = 0; j < 8; ++j)
            H1s[(moff + j) * H1_LD + ncol] = fmaxf(acc[j] + bias, 0.f);
    }
    __syncthreads();

    // ---------------- Layer 2: [16x256] @ W2[256x64] + b2, ReLU -> H2s ----------
    for (int nt = 0; nt < N2 / 16; ++nt) {
        const int ncol = nt * 16 + lrow;
        v8f acc = {0.f, 0.f, 0.f, 0.f, 0.f, 0.f, 0.f, 0.f};
        for (int kk = 0; kk < N1; kk += 4) {
            v2f a, bm;
            a.x  = H1s[lrow * H1_LD + kk + khalf];
            a.y  = H1s[lrow * H1_LD + kk + khalf + 1];
            bm.x = W2[(kk + khalf)     * N2 + ncol];
            bm.y = W2[(kk + khalf + 1) * N2 + ncol];
            acc = __builtin_amdgcn_wmma_f32_16x16x4_f32(
                false, a, false, bm, (short)0, acc, false, false);
        }
        const float bias = b2[ncol];
        #pragma unroll
        for (int j = 0; j < 8; ++j)
            H2s[(moff + j) * H2_LD + ncol] = fmaxf(acc[j] + bias, 0.f);
    }
    __syncthreads();

    // ---------------- Layer 3: [16x64] @ W3[64x16] + b3 -> voxel_scores ---------
    {
        v8f acc = {0.f, 0.f, 0.f, 0.f, 0.f, 0.f, 0.f, 0.f};
        for (int kk = 0; kk < N2; kk += 4) {
            v2f a, bm;
            a.x  = H2s[lrow * H2_LD + kk + khalf];
            a.y  = H2s[lrow * H2_LD + kk + khalf + 1];
            bm.x = W3[(kk + khalf)     * N3 + lrow];
            bm.y = W3[(kk + khalf + 1) * N3 + lrow];
            acc = __builtin_amdgcn_wmma_f32_16x16x4_f32(
                false, a, false, bm, (short)0, acc, false, false);
        }
        const float bias = b3[lrow];
        #pragma unroll
        for (int j = 0; j < 8; ++j)
            voxel_scores[(size_t)(rbase + moff + j) * 16 + lrow] = acc[j] + bias;
    }
}

// ---------------- Stage 3: per-point score gather ------------------------------
__global__ void gather_point_scores_kernel(
    const float* __restrict__ voxel_scores,   // (B,V,16)
    const int*   __restrict__ point2voxel,    // (B,P)
    float*       __restrict__ out)            // (B,P,16)
{
    const int gid = blockIdx.x * blockDim.x + threadIdx.x;
    if (gid >= Bn * Pn * 16) return;
    const int n  = gid & 15;
    const int bp = gid >> 4;
    const int b  = bp / Pn;
    const int p  = bp - b * Pn;
    const int v  = point2voxel[b * Pn + p];
    out[gid] = voxel_scores[((size_t)(b * Vn + v)) * 16 + n];
}

extern "C" void kernel_launch(void* const* d_in, const int* in_sizes, int n_in,
                              void* d_out, int out_size, void* d_ws, size_t ws_size,
                              hipStream_t stream) {
    const float* patch_tokens   = (const float*)d_in[0];
    const float* voxel_features = (const float*)d_in[1];
    const float* voxel_coords   = (const float*)d_in[2];
    const float* image_sizes    = (const float*)d_in[3];
    const float* cam_intrinsics = (const float*)d_in[4];
    const float* lidar2cam      = (const float*)d_in[5];
    const int*   point2voxel    = (const int*)d_in[6];
    const float* W1 = (const float*)d_in[7];
    const float* b1 = (const float*)d_in[8];
    const float* W2 = (const float*)d_in[9];
    const float* b2 = (const float*)d_in[10];
    const float* W3 = (const float*)d_in[11];
    const float* b3 = (const float*)d_in[12];

    float* out = (float*)d_out;
    float* voxel_scores = (float*)d_ws;   // needs B*V*16*4 = 6.4 MB of scratch

    // 100000 fused rows / 16-row tiles = 6250 waves, one wave per block
    fused_proj_mlp_kernel<<<(Bn * Vn) / 16, 32, 0, stream>>>(
        patch_tokens, voxel_features, voxel_coords, image_sizes,
        cam_intrinsics, lidar2cam, W1, b1, W2, b2, W3, b3, voxel_scores);

    const int total = Bn * Pn * 16;
    gather_point_scores_kernel<<<(total + 255) / 256, 256, 0, stream>>>(
        voxel_scores, point2voxel, out);
}